// DeepSetsAggregator_50010599195089
// MI455X (gfx1250) — compile-verified
//
#include <hip/hip_runtime.h>
#include <stdint.h>

typedef __attribute__((ext_vector_type(2))) float v2f;
typedef __attribute__((ext_vector_type(8))) float v8f;

#define HIDDEN 128
#define NTILE 8   // 128 features / 16 per WMMA tile

// One wave32 per segment. Sorted batch ids -> contiguous row range per segment.
// Sum via V_WMMA_F32_16X16X4_F32 with A=ones (matrix pipe), max via VALU.
// Hot loop is branch-free & select-free so loads batch into one burst per
// iteration (32 outstanding b32 loads) instead of per-load exec-masked waits.
__global__ __launch_bounds__(32) void deepsets_seg_pool(
    const float* __restrict__ x,
    const long long* __restrict__ batch,   // int64 sorted segment ids
    float* __restrict__ out,               // [B, 3*HIDDEN] : sum | mean | max
    int n, int bsz)
{
  const int b = blockIdx.x;
  if (b >= bsz) return;

  // lower_bound(batch, b) and lower_bound(batch, b+1) — uniform across lanes.
  int lo = 0, hi = n;
  while (lo < hi) { int mid = (lo + hi) >> 1; if (batch[mid] < (long long)b) lo = mid + 1; else hi = mid; }
  const int start = lo;
  hi = n;
  while (lo < hi) { int mid = (lo + hi) >> 1; if (batch[mid] < (long long)(b + 1)) lo = mid + 1; else hi = mid; }
  const int count = lo - start;

  const int lane = threadIdx.x;   // 0..31
  const int n16  = lane & 15;     // feature within 16-wide tile (column N)
  const int h    = lane >> 4;     // half-wave id

  const float NEG_INF = -__builtin_inff();

  v8f  acc[NTILE];
  float mx[NTILE];
  const v8f zero = {};
#pragma unroll
  for (int t = 0; t < NTILE; ++t) { acc[t] = zero; mx[t] = NEG_INF; }

  // A = all-ones 16x4 f32 (2 VGPRs per lane) -> D[m][n] += sum_k B[k][n]
  v2f onesA; onesA.x = 1.0f; onesA.y = 1.0f;

  const float* xrow = x + (size_t)start * HIDDEN;

  const int rmain = count & ~7;   // rows handled by the unconditional hot loop

  for (int r = 0; r < rmain; r += 8) {
    // Prefetch next 8-row window (4 KB): 32 lanes x 128 B.
    if (r + 8 < count)
      __builtin_prefetch(xrow + (size_t)(r + 8) * HIDDEN + lane * 32, 0, 1);

    // ---- load phase: 32 independent unconditional loads, no consumers ----
    float va[2][NTILE];  // rows r + g*4 + h
    float vb[2][NTILE];  // rows r + g*4 + 2 + h
#pragma unroll
    for (int g = 0; g < 2; ++g) {
      const float* p0 = xrow + (size_t)(r + g * 4 + h)     * HIDDEN + n16;
      const float* p1 = xrow + (size_t)(r + g * 4 + 2 + h) * HIDDEN + n16;
#pragma unroll
      for (int t = 0; t < NTILE; ++t) {
        va[g][t] = p0[t * 16];
        vb[g][t] = p1[t * 16];
      }
    }

    // ---- consume phase: 16 WMMAs on the matrix pipe + VALU max ----
#pragma unroll
    for (int g = 0; g < 2; ++g) {
#pragma unroll
      for (int t = 0; t < NTILE; ++t) {
        v2f Bm; Bm.x = va[g][t]; Bm.y = vb[g][t];
        acc[t] = __builtin_amdgcn_wmma_f32_16x16x4_f32(
            false, onesA, false, Bm, (short)0, acc[t], false, false);
        mx[t] = fmaxf(mx[t], fmaxf(va[g][t], vb[g][t]));
      }
    }
  }

  // ---- tail: up to 7 rows; clamp row index so loads stay unconditional ----
  if (rmain < count) {
    const int last = count - 1;   // count >= 1 here
#pragma unroll
    for (int g = 0; g < 2; ++g) {
      const int r0 = rmain + g * 4 + h;
      const int r1 = rmain + g * 4 + 2 + h;
      const int c0 = r0 <= last ? r0 : last;
      const int c1 = r1 <= last ? r1 : last;
      const float* p0 = xrow + (size_t)c0 * HIDDEN + n16;
      const float* p1 = xrow + (size_t)c1 * HIDDEN + n16;
      float w0[NTILE], w1[NTILE];
#pragma unroll
      for (int t = 0; t < NTILE; ++t) { w0[t] = p0[t * 16]; w1[t] = p1[t * 16]; }
#pragma unroll
      for (int t = 0; t < NTILE; ++t) {
        const float v0 = (r0 < count) ? w0[t] : 0.0f;
        const float v1 = (r1 < count) ? w1[t] : 0.0f;
        v2f Bm; Bm.x = v0; Bm.y = v1;
        acc[t] = __builtin_amdgcn_wmma_f32_16x16x4_f32(
            false, onesA, false, Bm, (short)0, acc[t], false, false);
        const float m0 = (r0 < count) ? w0[t] : NEG_INF;
        const float m1 = (r1 < count) ? w1[t] : NEG_INF;
        mx[t] = fmaxf(mx[t], fmaxf(m0, m1));
      }
    }
  }

  const float inv = 1.0f / (float)(count > 0 ? count : 1);
  float* ob = out + (size_t)b * (3 * HIDDEN);

#pragma unroll
  for (int t = 0; t < NTILE; ++t) {
    // acc[t][0]: lanes 0-15 hold D[0][n], lanes 16-31 hold D[8][n];
    // with A=ones both equal the total per-feature sum.
    const float s = acc[t][0];
    float m = fmaxf(mx[t], __shfl_xor(mx[t], 16, 32));  // combine half-waves
    m = (count > 0) ? m : 0.0f;
    const int col = t * 16 + n16;
    if (h == 0) {
      ob[col]          = s;        // sum
      ob[HIDDEN + col] = s * inv;  // mean
    } else {
      ob[2 * HIDDEN + col] = m;    // max (0 for empty segments)
    }
  }
}

extern "C" void kernel_launch(void* const* d_in, const int* in_sizes, int n_in,
                              void* d_out, int out_size, void* d_ws, size_t ws_size,
                              hipStream_t stream) {
  (void)n_in; (void)d_ws; (void)ws_size;
  const float*     x     = (const float*)d_in[0];
  const long long* batch = (const long long*)d_in[1];  // int64 per reference
  float*           out   = (float*)d_out;

  const int n   = in_sizes[1];               // number of rows
  const int bsz = out_size / (3 * HIDDEN);   // B derived from output shape

  deepsets_seg_pool<<<bsz, 32, 0, stream>>>(x, batch, out, n, bsz);
}